// EdgeIndexGenerator_12524124635757
// MI455X (gfx1250) — compile-verified
//
#include <hip/hip_runtime.h>
#include <hip/hip_bf16.h>

// Edge-index generation for a B x H x W 4-neighbor grid graph (reference:
// EdgeIndexGenerator). Output layout: d_out[0..E-1] = source row,
// d_out[E..2E-1] = destination row (int32, matching jax default int).
//
// Zero FLOPs -> WMMA/TDM inapplicable; this is a pure store-bandwidth kernel
// (~14.2 MB of int32 stores -> ~0.6 us at 23.3 TB/s HBM). One thread per node,
// closed-form exclusive prefix of per-node edge counts (no scan / atomics),
// contiguous per-node slots so wave32 stores coalesce.
//
// All index math fits in uint32 (out_size is a host int, so E < 2^31 and every
// offset/node id < E): this avoids the huge 64-bit software-division
// expansions that dominated the previous build's SALU/VALU counts.

__global__ void __launch_bounds__(256)
edge_index_gen_kernel(const int* __restrict__ hptr,
                      const int* __restrict__ wptr,
                      int* __restrict__ out,
                      unsigned int E) {
  const unsigned int H = (unsigned int)*hptr;   // uniform -> scalar load
  const unsigned int W = (unsigned int)*wptr;
  const unsigned int HW = H * W;
  if (HW == 0u) return;

  // Directed edges contributed by one batch image (fits in u32: P <= 9*HW <= 9E).
  const unsigned int P = HW + 4u * H * (W - 1u) + 4u * W * (H - 1u);

  // Recover batch count + whether the last-node symmetric pair was appended.
  // (E % P == 0 only in the H*W == 1 degenerate case, where no pair is added.)
  unsigned int B, extra;
  if ((E % P) == 0u) { B = E / P;       extra = 0u; }
  else               { B = (E - 2u) / P; extra = 2u; }
  const unsigned int N = B * HW;

  const unsigned int i = blockIdx.x * blockDim.x + threadIdx.x;
  if (i >= N) return;

  const unsigned int b = i / HW;        // short u32 VALU division expansion
  const unsigned int p = i - b * HW;
  const unsigned int y = p / W;
  const unsigned int x = p - y * W;

  const unsigned int hasL = (x > 0u);
  const unsigned int hasR = (x < W - 1u);
  const unsigned int hasU = (y > 0u);
  const unsigned int hasD = (y < H - 1u);
  const unsigned int vert = hasU + hasD;

  // Exclusive prefix of edge counts over all nodes before i.
  unsigned int off = b * P
                   + y * (5u * W - 4u)                      // base row sums
                   + 2u * W * ((hasU ? (y - 1u) : 0u) + y)  // up/down in prior rows
                   + x                                      // selfs in this row
                   + 2u * (hasL ? (x - 1u) : 0u)            // left pairs before x
                   + 2u * x                                 // right pairs before x
                   + 2u * x * vert;                         // up/down before x

  int* __restrict__ src = out;
  int* __restrict__ dst = out + E;

  const int ii = (int)i;
  unsigned int pos = off;

  // self edge
  src[pos] = ii; dst[pos] = ii; ++pos;

  if (hasL) {                        // left pair
    const int l = ii - 1;
    src[pos] = ii; dst[pos] = l;  ++pos;
    src[pos] = l;  dst[pos] = ii; ++pos;
  }
  if (hasR) {                        // right pair
    const int r = ii + 1;
    src[pos] = ii; dst[pos] = r;  ++pos;
    src[pos] = r;  dst[pos] = ii; ++pos;
  }
  if (hasU) {                        // up pair
    const int u = ii - (int)W;
    src[pos] = ii; dst[pos] = u;  ++pos;
    src[pos] = u;  dst[pos] = ii; ++pos;
  }
  if (hasD) {                        // down pair
    const int d = ii + (int)W;
    src[pos] = ii; dst[pos] = d;  ++pos;
    src[pos] = d;  dst[pos] = ii; ++pos;
  }

  // Loop-variable leak in the original torch code: one symmetric pair for the
  // LAST node only. Last node has x = W-1, y = H-1 -> mirror is (0,0) of the
  // same batch image.
  if (extra && i == N - 1u) {
    const unsigned int sidx = b * HW;  // sy = sx = 0
    src[E - 2u] = ii;          dst[E - 2u] = (int)sidx;
    src[E - 1u] = (int)sidx;   dst[E - 1u] = ii;
  }
}

extern "C" void kernel_launch(void* const* d_in, const int* in_sizes, int n_in,
                              void* d_out, int out_size, void* d_ws, size_t ws_size,
                              hipStream_t stream) {
  (void)in_sizes; (void)n_in; (void)d_ws; (void)ws_size;

  // Inputs (setup_inputs order): [0] node_features f32 (unused — only its
  // length matters and that is implied by out_size), [1] height (int scalar,
  // device), [2] width (int scalar, device).
  const int* hptr = (const int*)d_in[1];
  const int* wptr = (const int*)d_in[2];
  int* out = (int*)d_out;

  const unsigned int E = (unsigned int)(out_size / 2);

  // N = B*HW <= E always (per-batch edges >= per-batch nodes), so a grid of
  // ceil(E/256) blocks of 256 threads is guaranteed to cover every node; excess threads exit.
  const int threads = 256;                       // 8 wave32 waves per block
  const unsigned int blocks = (E + threads - 1) / threads;

  edge_index_gen_kernel<<<(blocks > 0u ? blocks : 1u), threads, 0, stream>>>(
      hptr, wptr, out, E);
}